// BatchSeparationLoss_83786222011133
// MI455X (gfx1250) — compile-verified
//
#include <hip/hip_runtime.h>

typedef __attribute__((ext_vector_type(2))) float v2f;
typedef __attribute__((ext_vector_type(8))) float v8f;

#define HW      50176      // 224*224
#define BATCH   16
#define NMAP    16
#define MAPS    256        // BATCH*NMAP
#define SLICES  28         // K-slices per batch
#define WAVES   8          // waves per block
#define KPW     224        // K values per wave  (28*8*224 == 50176)
#define STEPS   56         // KPW / 4

// ---------------- Pass 1: per-map min/max -> (mn, 1/(range+eps)) ----------------
__global__ __launch_bounds__(256)
void bsl_minmax_kernel(const float* __restrict__ h, float2* __restrict__ stats) {
    __shared__ float smn[256];
    __shared__ float smx[256];
    const int map = blockIdx.x;
    const float* p = h + (size_t)map * HW;
    float mn = 3.402823466e38f, mx = -3.402823466e38f;
    for (int i = threadIdx.x; i < HW; i += 256) {   // 196 iterations, coalesced
        float v = p[i];
        mn = fminf(mn, v);
        mx = fmaxf(mx, v);
    }
    smn[threadIdx.x] = mn;
    smx[threadIdx.x] = mx;
    __syncthreads();
    for (int off = 128; off > 0; off >>= 1) {
        if (threadIdx.x < off) {
            smn[threadIdx.x] = fminf(smn[threadIdx.x], smn[threadIdx.x + off]);
            smx[threadIdx.x] = fmaxf(smx[threadIdx.x], smx[threadIdx.x + off]);
        }
        __syncthreads();
    }
    if (threadIdx.x == 0) {
        float lo = smn[0], hi = smx[0];
        stats[map] = make_float2(lo, 1.0f / (hi - lo + 1e-8f));
    }
}

// ---------------- Pass 2: Gram via V_WMMA_F32_16X16X4_F32, tril-masked partial sums ----
__global__ __launch_bounds__(256)
void bsl_gram_kernel(const float* __restrict__ h,
                     const float2* __restrict__ stats,
                     float* __restrict__ partial) {
    __shared__ float red[WAVES];
    const int b     = blockIdx.x / SLICES;
    const int slice = blockIdx.x % SLICES;
    const int wave  = threadIdx.x >> 5;
    const int lane  = threadIdx.x & 31;
    const int m     = lane & 15;      // A-row index == B-col index for this lane
    const int hi    = lane >> 4;      // lane-half selects K pair {0,1} vs {2,3}

    const float* row = h + (size_t)(b * NMAP + m) * HW;
    const float2 st  = stats[b * NMAP + m];
    const float mn = st.x, inv = st.y;

    const int k0 = (slice * WAVES + wave) * KPW + (hi << 1);

    v8f acc = {};
    #pragma unroll 4
    for (int s = 0; s < STEPS; ++s) {
        const int k = k0 + s * 4;                       // even -> 8B aligned
        const float2 d = *(const float2*)(row + k);     // global_load_b64, streaming
        v2f a;
        a.x = (d.x - mn) * inv;
        a.y = (d.y - mn) * inv;
        // D = A*B + C, A-tile and B-tile of H*H^T share identical lane layout:
        // lane L holds h'[L&15][kbase + 2*(L>>4) + {0,1}] for both operands.
        acc = __builtin_amdgcn_wmma_f32_16x16x4_f32(
            /*neg_a=*/false, a, /*neg_b=*/false, a,
            /*c_mod=*/(short)0, acc, /*reuse_a=*/false, /*reuse_b=*/false);
    }

    // C/D layout: VGPR v holds gram[M][N] with M = v + 8*hi, N = m.
    // Keep strict lower triangle M > N.
    float s = 0.0f;
    #pragma unroll
    for (int v = 0; v < 8; ++v) {
        const int M = v + 8 * hi;
        if (M > m) s += acc[v];
    }
    // wave32 reduce
    for (int off = 16; off > 0; off >>= 1) s += __shfl_down(s, off, 32);
    if (lane == 0) red[wave] = s;
    __syncthreads();
    if (threadIdx.x == 0) {
        float t = 0.0f;
        #pragma unroll
        for (int w = 0; w < WAVES; ++w) t += red[w];
        partial[blockIdx.x] = t;
    }
}

// ---------------- Pass 3: deterministic final reduction, divide by B ----------------
__global__ __launch_bounds__(512)
void bsl_reduce_kernel(const float* __restrict__ partial, float* __restrict__ out, int n) {
    __shared__ float sm[512];
    float s = (threadIdx.x < (unsigned)n) ? partial[threadIdx.x] : 0.0f;
    sm[threadIdx.x] = s;
    __syncthreads();
    for (int off = 256; off > 0; off >>= 1) {
        if (threadIdx.x < off) sm[threadIdx.x] += sm[threadIdx.x + off];
        __syncthreads();
    }
    if (threadIdx.x == 0) out[0] = sm[0] / (float)BATCH;
}

extern "C" void kernel_launch(void* const* d_in, const int* in_sizes, int n_in,
                              void* d_out, int out_size, void* d_ws, size_t ws_size,
                              hipStream_t stream) {
    (void)in_sizes; (void)n_in; (void)out_size; (void)ws_size;
    const float* h = (const float*)d_in[0];
    float*       out = (float*)d_out;

    // workspace: [0..255] float2 stats (2KB), then 448 partial floats
    float2* stats   = (float2*)d_ws;
    float*  partial = (float*)d_ws + 2 * MAPS;

    bsl_minmax_kernel<<<MAPS, 256, 0, stream>>>(h, stats);
    bsl_gram_kernel<<<BATCH * SLICES, 256, 0, stream>>>(h, stats, partial);
    bsl_reduce_kernel<<<1, 512, 0, stream>>>(partial, out, BATCH * SLICES);
}